// ACT_R_31585189494803
// MI455X (gfx1250) — compile-verified
//
#include <hip/hip_runtime.h>
#include <cstdint>

#define SEQ_LEN 512
#define BATCH   256
#define NWAVE   8     // columns (waves) per block
#define WAVE    32    // gfx1250 is wave32

// v_log_f32 is log2; v_exp_f32 is exp2.
#define LOG2F(x) __builtin_amdgcn_logf(x)
#define EXP2F(x) __builtin_amdgcn_exp2f(x)

// One wave32 per batch column. History {scaled_t, p} lives in LDS as float2:
// one ds_load_b64 per (i,j) term, bank-conflict-free (32 lanes * 8B = 64 banks).
__global__ __launch_bounds__(NWAVE * WAVE, 1)
void act_r_scan_kernel(const float* __restrict__ sp,
                       const float* __restrict__ w,
                       float* __restrict__ out)
{
    __shared__ float2 scp[NWAVE][SEQ_LEN];   // .x = t_j * 86400*h, .y = p_j

    const int lane = threadIdx.x & (WAVE - 1);
    const int wid  = threadIdx.x >> 5;
    const int col  = blockIdx.x * NWAVE + wid;

    const float a   = w[0];
    const float c   = w[1];
    const float s   = w[2];
    const float tau = w[3];
    const float h   = w[4];
    const float K     = 86400.0f * h;
    const float inv_s = __builtin_amdgcn_rcpf(s);

    // ---- Stage sp[:, col] into LDS with gfx1250 async global->LDS copies ----
    // dsaddr = per-lane VGPR LDS byte address (low 32 bits of generic pointer),
    // tracked by ASYNCcnt (CDNA5 ISA).
    #pragma unroll
    for (int k = 0; k < SEQ_LEN / WAVE; ++k) {
        const int j = k * WAVE + lane;
        const float* g = sp + (size_t)j * BATCH + col;
        uint32_t ldsoff = (uint32_t)(uintptr_t)(&scp[wid][j].x);
        asm volatile("global_load_async_to_lds_b32 %0, %1, off"
                     :: "v"(ldsoff), "v"(g)
                     : "memory");
    }
    asm volatile("s_wait_asynccnt 0" ::: "memory");

    // Scale in place; zero-init p so masked-lane garbage can never appear.
    #pragma unroll
    for (int k = 0; k < SEQ_LEN / WAVE; ++k) {
        const int j = k * WAVE + lane;
        float x = scp[wid][j].x;
        scp[wid][j] = make_float2(x * K, 0.0f);
    }
    if (lane == 0) scp[wid][0].y = -a;   // p_0 = -(c*0 + a); exp(m_0)=0

    // ---- Sequential scan over i; inner j-sum parallel across 32 lanes ----
    for (int i = 1; i < SEQ_LEN; ++i) {
        const float sci = scp[wid][i].x;       // same address all lanes -> LDS broadcast
        const int nfull = i >> 5;
        float acc = 0.0f;

        #pragma unroll 4
        for (int cc = 0; cc < nfull; ++cc) {   // full 32-wide chunks, no mask needed
            float2 v = scp[wid][cc * WAVE + lane];
            float d  = fmaxf(sci - v.x, 1.0f);
            acc += EXP2F(v.y * LOG2F(d));
        }
        {   // remainder chunk, branchless mask (v_cndmask)
            const int jb = nfull << 5;
            float2 v = scp[wid][jb + lane];
            float d  = fmaxf(sci - v.x, 1.0f);
            float t  = EXP2F(v.y * LOG2F(d));
            acc += (jb + lane < i) ? t : 0.0f;
        }

        // wave32 butterfly reduction -> every lane holds S_i
        #pragma unroll
        for (int off = 16; off > 0; off >>= 1)
            acc += __shfl_xor(acc, off, WAVE);

        // exp(m_i) == S_i exactly, so p_i needs no log/exp round-trip.
        const float pi = -fmaf(c, acc, a);
        const float m  = LOG2F(acc) * 0.69314718055994530942f;   // ln(S_i)
        const float z  = (tau - m) * inv_s;
        const float e  = EXP2F(z * 1.4426950408889634f);
        const float act = __builtin_amdgcn_rcpf(1.0f + e);

        if (lane == 0) {
            scp[wid][i].y = pi;                          // consumed by steps > i (LDS in-order per wave)
            out[(size_t)(i - 1) * BATCH + col] = act;    // output is m[1:] activated
        }
    }
}

extern "C" void kernel_launch(void* const* d_in, const int* in_sizes, int n_in,
                              void* d_out, int out_size, void* d_ws, size_t ws_size,
                              hipStream_t stream) {
    (void)in_sizes; (void)n_in; (void)out_size; (void)d_ws; (void)ws_size;
    const float* sp = (const float*)d_in[0];   // [512, 256, 1] float32
    const float* w  = (const float*)d_in[1];   // [5] float32
    float* out = (float*)d_out;                // [511, 256, 1] float32

    dim3 grid(BATCH / NWAVE);                  // 32 blocks
    dim3 block(NWAVE * WAVE);                  // 256 threads = 8 wave32
    act_r_scan_kernel<<<grid, block, 0, stream>>>(sp, w, out);
}